// RIN_53523882443301
// MI455X (gfx1250) — compile-verified
//
#include <hip/hip_runtime.h>
#include <hip/hip_bf16.h>
#include <math.h>

#define BB 4096
#define TT 48
#define FF 35
#define HH 108
#define TF (TT * FF)

typedef __attribute__((ext_vector_type(2))) float v2f;
typedef __attribute__((ext_vector_type(8))) float v8f;

// Explicit global address-space pointers so the backend emits global_load/
// global_store (pure LOADcnt/STOREcnt) instead of flat ops (which also tie up
// DScnt and couple with the LDS pipeline feeding the WMMAs).
typedef const float __attribute__((address_space(1))) * gcfp;
typedef float __attribute__((address_space(1))) * gfp;

// d_out layout (flat, return order): x_imp, y_out, y_score, x_loss, xus, xrs
#define NXIMP   ((size_t)BB * TT * FF)
#define OFF_XIMP ((size_t)0)
#define OFF_YOUT (NXIMP)
#define OFF_YSC  (NXIMP + (size_t)BB)
#define OFF_LOSS (NXIMP + (size_t)2 * BB)
#define OFF_XUS  (NXIMP + (size_t)2 * BB + 1)
#define OFF_XRS  (OFF_XUS + NXIMP)

#define V8Z ((v8f){0.f, 0.f, 0.f, 0.f, 0.f, 0.f, 0.f, 0.f})

// ---------------- WMMA helper (V_WMMA_F32_16X16X4_F32, fp32 in/out) ----------
__device__ __forceinline__ v8f wmma4(v2f a, v2f b, v8f c) {
  // 8 args: (neg_a, A, neg_b, B, c_mod, C, reuse_a, reuse_b)
  return __builtin_amdgcn_wmma_f32_16x16x4_f32(false, a, false, b, (short)0, c,
                                               false, false);
}

__device__ __forceinline__ float sigf(float v) { return 1.0f / (1.0f + expf(-v)); }

// Opaque pointer laundering: blocks LICM from hoisting weight-fragment loads
// out of the time loop (the cause of the round-1 scratch spills).
__device__ __forceinline__ gcfp opaque(const float* p) {
  size_t v = (size_t)p;
  asm volatile("" : "+v"(v));
  return (gcfp)(const float*)v;
}

// A fragment (16x4, f32): lane l holds row = l&15, K = k0 + 2*(l>=16) + {0,1}
// Branch-free: clamped load + select.
__device__ __forceinline__ v2f a_frag_g(gcfp rowbase, int row_stride, int k0,
                                        int kmax, int lane) {
  int r = lane & 15;
  int kb = k0 + ((lane >> 4) << 1);
  gcfp p = rowbase + (size_t)r * row_stride;
  float v0 = p[min(kb + 0, kmax - 1)];
  float v1 = p[min(kb + 1, kmax - 1)];
  v2f a;
  a.x = (kb + 0 < kmax) ? v0 : 0.0f;
  a.y = (kb + 1 < kmax) ? v1 : 0.0f;
  return a;
}

// A fragment of xbar = m*x + (1-m)*x_hat built on the fly from global
__device__ __forceinline__ v2f a_frag_xbar(gcfp x, gcfp xh, gcfp m, size_t base,
                                           int row_stride, int k0, int lane) {
  int r = lane & 15;
  int kb = k0 + ((lane >> 4) << 1);
  size_t p = base + (size_t)r * row_stride;
  int k0c = min(kb + 0, FF - 1), k1c = min(kb + 1, FF - 1);
  float m0 = m[p + k0c], m1 = m[p + k1c];
  float v0 = m0 * x[p + k0c] + (1.0f - m0) * xh[p + k0c];
  float v1 = m1 * x[p + k1c] + (1.0f - m1) * xh[p + k1c];
  v2f a;
  a.x = (kb + 0 < FF) ? v0 : 0.0f;
  a.y = (kb + 1 < FF) ? v1 : 0.0f;
  return a;
}

// A fragment from LDS (row-major, given stride; pad columns pre-zeroed)
__device__ __forceinline__ v2f a_frag_l(const float* buf, int stride, int k0,
                                        int lane) {
  int r = lane & 15;
  int kb = k0 + ((lane >> 4) << 1);
  v2f a;
  a.x = buf[r * stride + kb];
  a.y = buf[r * stride + kb + 1];
  return a;
}

// B fragment (4x16): lane l holds col = l&15, K = k0 + 2*(l>=16) + {0,1}.
// B[k][n] = W[n][k] for W stored [Ntot][Ktot] row-major (i.e. X @ W^T).
// offdiag: zero the diagonal (FeatureRegression mask). Branch-free.
__device__ __forceinline__ v2f b_frag_w(gcfp W, int Ntot, int Ktot, int n0,
                                        int k0, bool offdiag, int lane) {
  int c = lane & 15;
  int kb = k0 + ((lane >> 4) << 1);
  int n = n0 + c;
  gcfp row = W + (size_t)min(n, Ntot - 1) * Ktot;
  float r0 = row[min(kb + 0, Ktot - 1)];
  float r1 = row[min(kb + 1, Ktot - 1)];
  bool ok0 = (n < Ntot) && (kb + 0 < Ktot) && !(offdiag && (kb + 0) == n);
  bool ok1 = (n < Ntot) && (kb + 1 < Ktot) && !(offdiag && (kb + 1) == n);
  v2f b;
  b.x = ok0 ? r0 : 0.0f;
  b.y = ok1 ? r1 : 0.0f;
  return b;
}

// ---------------- kernels ----------------------------------------------------
extern "C" __global__ void brits_zero(float* wsf) {
  int i = threadIdx.x;
  if (i < 2 * TT) wsf[i] = 0.0f;
}

extern "C" __global__ __launch_bounds__(32) void brits_main(
    const float* __restrict__ x_, const float* __restrict__ xhat_,
    const float* __restrict__ u_, const float* __restrict__ m_,
    const float* __restrict__ d_,
    const float* __restrict__ W_dec, const float* __restrict__ b_dec_,
    const float* __restrict__ W_hist, const float* __restrict__ b_hist_,
    const float* __restrict__ W_v, const float* __restrict__ b_v_,
    const float* __restrict__ W_r, const float* __restrict__ b_r_,
    const float* __restrict__ W_unc, const float* __restrict__ b_unc_,
    const float* __restrict__ conv_w, const float* __restrict__ conv_b,
    const float* __restrict__ W_ih, const float* __restrict__ W_hh,
    const float* __restrict__ b_ih_, const float* __restrict__ b_hh_,
    const float* __restrict__ W_out_, const float* __restrict__ b_out_,
    float* __restrict__ out_, float* __restrict__ ws_num,
    float* __restrict__ ws_den) {
  const int lane = threadIdx.x & 31;
  const int col  = lane & 15;   // N coordinate of this lane in C-layout
  const int half = lane >> 4;   // 0: M in 0..7, 1: M in 8..15
  const int b0   = blockIdx.x * 16;

  // Global-AS views of everything we dereference in the hot loop.
  gcfp x = (gcfp)x_, xhat = (gcfp)xhat_, u = (gcfp)u_, m = (gcfp)m_, d = (gcfp)d_;
  gcfp b_dec = (gcfp)b_dec_, b_hist = (gcfp)b_hist_, b_v = (gcfp)b_v_;
  gcfp b_r = (gcfp)b_r_, b_unc = (gcfp)b_unc_, b_ih = (gcfp)b_ih_;
  gcfp b_hh = (gcfp)b_hh_, W_out = (gcfp)W_out_, b_out = (gcfp)b_out_;
  gfp out = (gfp)out_;

  __shared__ float hbufA[16 * 112];  // hidden state ping (K-pad 108->112)
  __shared__ float hbufB[16 * 112];  // hidden state pong
  __shared__ float inpbuf[16 * 72];  // [x_imp | m] GRU input (70->72)
  __shared__ float xrbuf[16 * 36];   // x_r staged (35->36)

  for (int i = lane; i < 16 * 112; i += 32) { hbufA[i] = 0.0f; hbufB[i] = 0.0f; }
  for (int i = lane; i < 16 * 72;  i += 32) inpbuf[i] = 0.0f;
  for (int i = lane; i < 16 * 36;  i += 32) xrbuf[i] = 0.0f;
  __syncthreads();

  const float cw0 = conv_w[0], cw1 = conv_w[1], cb = conv_b[0];

  for (int t = 0; t < TT; ++t) {
    const size_t base = ((size_t)b0 * TT + t) * FF;  // element (b0, t, 0)
    float* hc = (t & 1) ? hbufB : hbufA;  // h entering this step
    float* hn = (t & 1) ? hbufA : hbufB;  // h leaving this step

    // Re-derive weight pointers per step so fragment loads cannot be hoisted
    // (and spilled) across the time loop.
    gcfp Wdec  = opaque(W_dec);
    gcfp Whist = opaque(W_hist);
    gcfp Wunc  = opaque(W_unc);
    gcfp Wv    = opaque(W_v);
    gcfp Wr    = opaque(W_r);
    gcfp Wih   = opaque(W_ih);
    gcfp Whh   = opaque(W_hh);

    // ---- Phase A: gamma_h = exp(-relu(d @ W_dec^T + b)); h *= gamma (in LDS)
#pragma unroll 1
    for (int nt = 0; nt < 7; ++nt) {
      v8f acc = V8Z;
      for (int k0 = 0; k0 < 36; k0 += 4) {
        v2f a  = a_frag_g(d + base, TF, k0, FF, lane);
        v2f bw = b_frag_w(Wdec, HH, FF, nt * 16, k0, false, lane);
        acc = wmma4(a, bw, acc);
      }
      int n = nt * 16 + col;
      float bd = (n < HH) ? b_dec[n] : 0.0f;
#pragma unroll
      for (int i = 0; i < 8; ++i) {
        float gam = expf(-fmaxf(acc[i] + bd, 0.0f));
        hc[(i + 8 * half) * 112 + n] *= gam;
      }
    }
    __syncthreads();

    // ---- Phase B: x_h = h @ W_hist^T + b; x_r = m*x + (1-m)*x_h -> LDS
#pragma unroll 1
    for (int ft = 0; ft < 3; ++ft) {
      v8f acc = V8Z;
      for (int k0 = 0; k0 < 112; k0 += 4) {
        v2f a  = a_frag_l(hc, 112, k0, lane);
        v2f bw = b_frag_w(Whist, FF, HH, ft * 16, k0, false, lane);
        acc = wmma4(a, bw, acc);
      }
      int n = ft * 16 + col;
      if (n < FF) {
        float bh = b_hist[n];
#pragma unroll
        for (int i = 0; i < 8; ++i) {
          int mr = i + 8 * half;
          size_t idx = base + (size_t)mr * TF + n;
          float xv = x[idx], mv = m[idx];
          float xh = acc[i] + bh;
          xrbuf[mr * 36 + n] = mv * xv + (1.0f - mv) * xh;
        }
      }
    }
    __syncthreads();

    // ---- Phase C: unc, xu, xr, x_comb, loss, x_imp, GRU-input staging
    float pnum = 0.0f, pden = 0.0f;
#pragma unroll 1
    for (int ft = 0; ft < 3; ++ft) {
      v8f acc_u = V8Z, acc_v = V8Z, acc_r = V8Z;
      for (int k0 = 0; k0 < 36; k0 += 4) {
        v2f au = a_frag_g(u + base, TF, k0, FF, lane);
        v2f bu = b_frag_w(Wunc, FF, FF, ft * 16, k0, false, lane);
        acc_u = wmma4(au, bu, acc_u);
        v2f ab = a_frag_xbar(x, xhat, m, base, TF, k0, lane);
        v2f bv = b_frag_w(Wv, FF, FF, ft * 16, k0, true, lane);
        acc_v = wmma4(ab, bv, acc_v);
        v2f ar = a_frag_l(xrbuf, 36, k0, lane);
        v2f br = b_frag_w(Wr, FF, FF, ft * 16, k0, true, lane);
        acc_r = wmma4(ar, br, acc_r);
      }
      int n = ft * 16 + col;
      if (n < FF) {
        float bun = b_unc[n], bvn = b_v[n], brn = b_r[n];
#pragma unroll
        for (int i = 0; i < 8; ++i) {
          int mr = i + 8 * half;
          size_t idx = base + (size_t)mr * TF + n;
          float xv = x[idx], mv = m[idx];
          float unc = expf(-fmaxf(acc_u[i] + bun, 0.0f));
          float xuv = (acc_v[i] + bvn) * unc;
          float xrv = acc_r[i] + brn;
          float xc = cw0 * xuv + cw1 * xrv + cb;
          float xi = mv * xv + (1.0f - mv) * xc;
          out[OFF_XUS + idx]  = xuv;
          out[OFF_XRS + idx]  = xrv;
          out[OFF_XIMP + idx] = xi;
          pnum += fabsf(xv - xc) * mv;
          pden += mv;
          inpbuf[mr * 72 + n]      = xi;
          inpbuf[mr * 72 + FF + n] = mv;
        }
      }
    }
    // wave-level reduction, single atomic pair per (tile, t)
#pragma unroll
    for (int off = 16; off > 0; off >>= 1) {
      pnum += __shfl_down(pnum, off, 32);
      pden += __shfl_down(pden, off, 32);
    }
    if (lane == 0) {
      atomicAdd(&ws_num[t], pnum);
      atomicAdd(&ws_den[t], pden);
    }
    __syncthreads();

    // ---- Phase D: GRU cell; reads hc, writes hn
#pragma unroll 1
    for (int nt = 0; nt < 7; ++nt) {
      v8f gi[3], gh[3];
#pragma unroll
      for (int g = 0; g < 3; ++g) {
        v8f ai = V8Z;
        for (int k0 = 0; k0 < 72; k0 += 4) {
          v2f a  = a_frag_l(inpbuf, 72, k0, lane);
          v2f bw = b_frag_w(Wih + (size_t)g * HH * (2 * FF), HH, 2 * FF,
                            nt * 16, k0, false, lane);
          ai = wmma4(a, bw, ai);
        }
        gi[g] = ai;
        v8f ah = V8Z;
        for (int k0 = 0; k0 < 112; k0 += 4) {
          v2f a  = a_frag_l(hc, 112, k0, lane);
          v2f bw = b_frag_w(Whh + (size_t)g * HH * HH, HH, HH,
                            nt * 16, k0, false, lane);
          ah = wmma4(a, bw, ah);
        }
        gh[g] = ah;
      }
      int n = nt * 16 + col;
      float bir = 0, biz = 0, bin = 0, bhr = 0, bhz = 0, bhn = 0;
      if (n < HH) {
        bir = b_ih[n]; biz = b_ih[HH + n]; bin = b_ih[2 * HH + n];
        bhr = b_hh[n]; bhz = b_hh[HH + n]; bhn = b_hh[2 * HH + n];
      }
#pragma unroll
      for (int i = 0; i < 8; ++i) {
        int mr = i + 8 * half;
        float hold = hc[mr * 112 + n];
        float r  = sigf(gi[0][i] + bir + gh[0][i] + bhr);
        float z  = sigf(gi[1][i] + biz + gh[1][i] + bhz);
        float nn = tanhf(gi[2][i] + bin + r * (gh[2][i] + bhn));
        hn[mr * 112 + n] = (1.0f - z) * nn + z * hold;
      }
    }
    __syncthreads();
  }

  // ---- y_out = h_fin @ W_out^T + b; y_score = sigmoid(y_out)
  // TT even -> final h is in hbufA.
  const float* hf = hbufA;
  if (lane < 16) {
    float acc = b_out[0];
    for (int k = 0; k < HH; ++k) acc += hf[lane * 112 + k] * W_out[k];
    out[OFF_YOUT + b0 + lane] = acc;
    out[OFF_YSC  + b0 + lane] = sigf(acc);
  }
}

extern "C" __global__ void brits_loss(const float* __restrict__ wsn,
                                      const float* __restrict__ wsd,
                                      float* __restrict__ out) {
  if (threadIdx.x == 0 && blockIdx.x == 0) {
    float s = 0.0f;
    for (int t = 0; t < TT; ++t) s += wsn[t] / (wsd[t] + 1e-5f);
    out[OFF_LOSS] = s;
  }
}

extern "C" void kernel_launch(void* const* d_in, const int* in_sizes, int n_in,
                              void* d_out, int out_size, void* d_ws,
                              size_t ws_size, hipStream_t stream) {
  (void)in_sizes; (void)n_in; (void)out_size; (void)ws_size;
  const float* x      = (const float*)d_in[0];
  const float* xhat   = (const float*)d_in[1];
  const float* u      = (const float*)d_in[2];
  const float* m      = (const float*)d_in[3];
  const float* d      = (const float*)d_in[4];
  const float* W_dec  = (const float*)d_in[5];
  const float* b_dec  = (const float*)d_in[6];
  const float* W_hist = (const float*)d_in[7];
  const float* b_hist = (const float*)d_in[8];
  const float* W_v    = (const float*)d_in[9];
  const float* b_v    = (const float*)d_in[10];
  const float* W_r    = (const float*)d_in[11];
  const float* b_r    = (const float*)d_in[12];
  const float* W_unc  = (const float*)d_in[13];
  const float* b_unc  = (const float*)d_in[14];
  const float* conv_w = (const float*)d_in[15];
  const float* conv_b = (const float*)d_in[16];
  const float* W_ih   = (const float*)d_in[17];
  const float* W_hh   = (const float*)d_in[18];
  const float* b_ih   = (const float*)d_in[19];
  const float* b_hh   = (const float*)d_in[20];
  const float* W_out  = (const float*)d_in[21];
  const float* b_out  = (const float*)d_in[22];

  float* out = (float*)d_out;
  float* wsn = (float*)d_ws;        // num[48]
  float* wsd = wsn + TT;            // den[48]

  brits_zero<<<1, 128, 0, stream>>>(wsn);
  brits_main<<<BB / 16, 32, 0, stream>>>(
      x, xhat, u, m, d, W_dec, b_dec, W_hist, b_hist, W_v, b_v, W_r, b_r,
      W_unc, b_unc, conv_w, conv_b, W_ih, W_hh, b_ih, b_hh, W_out, b_out,
      out, wsn, wsd);
  brits_loss<<<1, 32, 0, stream>>>(wsn, wsd, out);
}